// custom2D_44401371906365
// MI455X (gfx1250) — compile-verified
//
#include <hip/hip_runtime.h>

// ---------------------------------------------------------------------------
// 3x3 "valid" cross-correlation, 8192x8192 f32 -> 8190x8190 f32, via
// V_WMMA_F32_16X16X4_F32 banded GEMM on gfx1250 (wave32).
//
// Per 16x16 output tile:  out = sum_{k=0..2} X_k(16x20) * W_k(20x16),
// W_k[c,n] = w[k, c-n] banded, split into 5 K=4 chunks => 15 f32 WMMAs.
// HBM-bound (~537 MB @ 23.3 TB/s => ~23 us floor).
//
// v6: round-5's hand-written asm loads were UNSAFE (scheduler interleaved
// WMMAs between them with no s_wait_loadcnt; allocator recycled one dest
// pair). Loads are compiler-visible again, but a single opaque asm barrier
// with all 15 fragments as "+v" operands separates the load batch from the
// WMMA chain: loads can't sink, WMMAs can't hoist, 15 values stay live =>
// 15 distinct dest pairs, one compiler-inserted s_wait_loadcnt 0x0.
// ---------------------------------------------------------------------------

typedef __attribute__((ext_vector_type(2))) float v2f;
typedef __attribute__((ext_vector_type(8))) float v8f;

#define IMG_W    8192
#define IMG_H    8192
#define OUT_W    8190
#define OUT_H    8190
#define TILES_X  512            // ceil(8190/16)
#define TILES_Y  512
#define T_PER_W  8              // tiles swept per wave (amortizes B setup)

__global__ __launch_bounds__(256) void conv3x3_wmma_f32(
    const float* __restrict__ letter,
    const float* __restrict__ kw,     // 9 floats, row-major 3x3
    float* __restrict__ out)
{
    const int lane = threadIdx.x & 31;                 // wave32
    const int wave = threadIdx.x >> 5;                 // 0..7
    const int tx   = blockIdx.x;                       // tile col 0..511
    const int ty0  = (blockIdx.y * 8 + wave) * T_PER_W;// first tile row
    const int j0   = tx * 16;

    const int mrow = lane & 15;   // M (A) / N (B,D) index of this lane
    const int hi   = lane >> 4;   // 0: K=0,1   1: K=2,3  (A & B fragments)

    // 3x3 weights: uniform -> scalar loads
    float w[3][3];
#pragma unroll
    for (int k = 0; k < 3; ++k)
#pragma unroll
        for (int l = 0; l < 3; ++l)
            w[k][l] = kw[k * 3 + l];

    // --- B fragments, built ONCE per wave, pinned in VGPRs. ----------------
    // B 4x16 f32 layout: lane n = lane%16; VGPR slot s holds K-row 2*hi+s.
    // Chunk ch covers band columns c = 4*ch + 2*hi + s; W_k[c,n] = w[k,c-n].
    v2f Bf[3][5];
#pragma unroll
    for (int k = 0; k < 3; ++k) {
#pragma unroll
        for (int ch = 0; ch < 5; ++ch) {
#pragma unroll
            for (int s = 0; s < 2; ++s) {
                const int c = 4 * ch + 2 * hi + s;
                const int d = c - mrow;
                float v = 0.0f;
                v = (d == 0) ? w[k][0] : v;
                v = (d == 1) ? w[k][1] : v;
                v = (d == 2) ? w[k][2] : v;
                Bf[k][ch][s] = v;
            }
            asm volatile("" : "+v"(Bf[k][ch]));   // pin: no remat in loop
        }
    }

    // Column element-offsets (clamped; fixed for the whole wave). Clamped
    // loads only corrupt band columns whose B entries are zero / output
    // columns >= 8190, which are never stored.
    unsigned colOff[5];
#pragma unroll
    for (int ch = 0; ch < 5; ++ch) {
        int c = j0 + 4 * ch + 2 * hi;                 // even -> 8B aligned
        c = (c < IMG_W - 2) ? c : (IMG_W - 2);
        colOff[ch] = (unsigned)c;
    }

    const int  ocol   = j0 + mrow;
    const bool edge_x = (tx == TILES_X - 1);          // wave-uniform

    for (int t = 0; t < T_PER_W; ++t) {
        const int ty = ty0 + t;
        const int i0 = ty * 16;

        // Row element-offsets (clamped; garbage rows never stored).
        unsigned rowOff[3];
#pragma unroll
        for (int k = 0; k < 3; ++k) {
            int row = i0 + mrow + k;
            row = (row < IMG_H - 1) ? row : (IMG_H - 1);
            rowOff[k] = (unsigned)row << 13;          // * IMG_W, < 2^26
        }

        // --- Stage ALL 15 A fragments (compiler-visible loads). -----------
        v2f a[3][5];
#pragma unroll
        for (int k = 0; k < 3; ++k)
#pragma unroll
            for (int ch = 0; ch < 5; ++ch)
                a[k][ch] = *(const v2f*)(letter + (rowOff[k] + colOff[ch]));

        // Scheduling fence: loads cannot sink below this point, WMMAs cannot
        // rise above it, and all 15 fragments are live here (=> 15 distinct
        // dest pairs, 15 loads in flight, one s_wait_loadcnt before this).
        asm volatile(""
            : "+v"(a[0][0]), "+v"(a[0][1]), "+v"(a[0][2]), "+v"(a[0][3]),
              "+v"(a[0][4]),
              "+v"(a[1][0]), "+v"(a[1][1]), "+v"(a[1][2]), "+v"(a[1][3]),
              "+v"(a[1][4]),
              "+v"(a[2][0]), "+v"(a[2][1]), "+v"(a[2][2]), "+v"(a[2][3]),
              "+v"(a[2][4]));

        // --- 15 chained WMMAs: acc += X_k_chunk(16x4) * W_k_chunk(4x16) ---
        v8f acc = {};
#pragma unroll
        for (int k = 0; k < 3; ++k)
#pragma unroll
            for (int ch = 0; ch < 5; ++ch) {
                // (neg_a, A, neg_b, B, c_mod, C, reuse_a, reuse_b)
                acc = __builtin_amdgcn_wmma_f32_16x16x4_f32(
                    false, a[k][ch], false, Bf[k][ch], (short)0, acc,
                    false, false);
            }

        // --- Store D: element (M = v + 8*hi, N = mrow), f32 C/D layout ----
        if (!edge_x && ty < TILES_Y - 1) {
            // interior: all 16 rows / 16 cols in range, no per-lane guards
            const unsigned base = (unsigned)(i0 + 8 * hi) * OUT_W + (unsigned)ocol;
#pragma unroll
            for (int v = 0; v < 8; ++v)
                out[(size_t)(base + (unsigned)v * OUT_W)] = acc[v];
        } else {
            if (ocol < OUT_W) {
#pragma unroll
                for (int v = 0; v < 8; ++v) {
                    const int orow = i0 + 8 * hi + v;
                    if (orow < OUT_H)
                        out[(size_t)orow * OUT_W + ocol] = acc[v];
                }
            }
        }
    }
}

extern "C" void kernel_launch(void* const* d_in, const int* in_sizes, int n_in,
                              void* d_out, int out_size, void* d_ws, size_t ws_size,
                              hipStream_t stream) {
    (void)in_sizes; (void)n_in; (void)out_size; (void)d_ws; (void)ws_size;
    const float* letter = (const float*)d_in[0];
    const float* kern   = (const float*)d_in[1];
    float* out          = (float*)d_out;

    // 8 waves/block; each wave sweeps 8 vertically consecutive tiles.
    dim3 grid(TILES_X, TILES_Y / (8 * T_PER_W) * 8, 1);   // (512, 64)
    dim3 block(256, 1, 1);
    hipLaunchKernelGGL(conv3x3_wmma_f32, grid, block, 0, stream,
                       letter, kern, out);
}